// GNN_11192684773414
// MI455X (gfx1250) — compile-verified
//
#include <hip/hip_runtime.h>

#define N_USERS 100000
#define N_ITEMS 100000
#define NBINS   100000      // rows per fold (users == items == 100000)
#define DIM     128
#define NNZV    1000000
#define ALPHA   0.2f
#define WLD     130         // LDS slot stride (floats): conflict-free paired b64 loads
#define SCANB   1024        // bins per scan block

typedef __attribute__((ext_vector_type(2))) float v2f;
typedef __attribute__((ext_vector_type(8))) float v8f;

// ===========================================================================
// CSR build: histogram -> exclusive scan -> scatter.
// Replaces 512M f32 atomics (L2 atomic-ALU bound, ~ms) with 8M u32 atomics
// plus a bandwidth-bound gather (~90us/fold at 23.3 TB/s).
// ===========================================================================
__global__ void gnn_zero_i(int* __restrict__ p, int n) {
    int i = blockIdx.x * blockDim.x + threadIdx.x;
    if (i < n) p[i] = 0;
}

__global__ void gnn_hist(const int* __restrict__ rows, int* __restrict__ counts,
                         int nnz) {
    int e = blockIdx.x * blockDim.x + threadIdx.x;
    if (e < nnz) atomicAdd(&counts[rows[e]], 1);
}

// Per-block exclusive scan of 1024 bins (256 thr x 4), emits block totals.
__global__ void gnn_scan_local(const int* __restrict__ counts,
                               int* __restrict__ offs,
                               int* __restrict__ bsums, int n) {
    __shared__ int sc[256];
    const int t  = threadIdx.x;
    const int i0 = blockIdx.x * SCANB + t * 4;
    int a0 = (i0 + 0 < n) ? counts[i0 + 0] : 0;
    int a1 = (i0 + 1 < n) ? counts[i0 + 1] : 0;
    int a2 = (i0 + 2 < n) ? counts[i0 + 2] : 0;
    int a3 = (i0 + 3 < n) ? counts[i0 + 3] : 0;
    int s = a0 + a1 + a2 + a3;
    sc[t] = s;
    __syncthreads();
    for (int d = 1; d < 256; d <<= 1) {            // Hillis-Steele inclusive
        int v = (t >= d) ? sc[t - d] : 0;
        __syncthreads();
        sc[t] += v;
        __syncthreads();
    }
    int excl = sc[t] - s;
    if (t == 255) bsums[blockIdx.x] = sc[255];
    if (i0 + 0 < n) offs[i0 + 0] = excl;
    if (i0 + 1 < n) offs[i0 + 1] = excl + a0;
    if (i0 + 2 < n) offs[i0 + 2] = excl + a0 + a1;
    if (i0 + 3 < n) offs[i0 + 3] = excl + a0 + a1 + a2;
}

// Single-block exclusive scan of the (<=128) block totals.
__global__ void gnn_scan_sums(int* __restrict__ bsums, int nb) {
    __shared__ int sc[128];
    const int t = threadIdx.x;
    int v = (t < nb) ? bsums[t] : 0;
    sc[t] = v;
    __syncthreads();
    for (int d = 1; d < 128; d <<= 1) {
        int u = (t >= d) ? sc[t - d] : 0;
        __syncthreads();
        sc[t] += u;
        __syncthreads();
    }
    if (t < nb) bsums[t] = sc[t] - v;              // exclusive
}

__global__ void gnn_scan_add(int* __restrict__ offs, const int* __restrict__ bsums,
                             int n) {
    int i = blockIdx.x * blockDim.x + threadIdx.x;
    if (i < n) offs[i] += bsums[i >> 10];
}

// Scatter edges into row-sorted col/val arrays (1 u32 atomic per edge).
__global__ void gnn_scatter(const int* __restrict__ rows, const int* __restrict__ cols,
                            const float* __restrict__ vals,
                            const int* __restrict__ offs, int* __restrict__ cursor,
                            int* __restrict__ scol, float* __restrict__ sval, int nnz) {
    int e = blockIdx.x * blockDim.x + threadIdx.x;
    if (e >= nnz) return;
    int r   = rows[e];
    int pos = atomicAdd(&cursor[r], 1);
    int dst = offs[r] + pos;
    scol[dst] = cols[e];
    sval[dst] = vals[e];
}

// Gather SpMM, ZERO atomics: one wave per output row, each lane owns a
// float4 chunk of the 128-dim row; ~10 edges/row (1M nnz / 100k rows).
// Per edge: one global_load_b128 per lane + FMA; one b128 store at the end.
__global__ void gnn_spmm_gather(const int* __restrict__ scol,
                                const float* __restrict__ sval,
                                const int* __restrict__ offs,
                                const int* __restrict__ counts,
                                const float* __restrict__ ebs,
                                float* __restrict__ acc, int n_rows) {
    int gid  = blockIdx.x * blockDim.x + threadIdx.x;
    int row  = gid >> 5;
    int lane = threadIdx.x & 31;
    if (row >= n_rows) return;
    int beg = offs[row];
    int cnt = counts[row];
    float4 a = make_float4(0.f, 0.f, 0.f, 0.f);
    for (int j = 0; j < cnt; ++j) {
        int   col = scol[beg + j];
        float v   = sval[beg + j];
        const float4 s = *(const float4*)(ebs + (size_t)col * DIM + (lane << 2));
        a.x += v * s.x; a.y += v * s.y; a.z += v * s.z; a.w += v * s.w;
    }
    *(float4*)(acc + (size_t)row * DIM + (lane << 2)) = a;   // all rows written
}

// ===========================================================================
// Fused dual GEMM + elementwise + leaky-relu on the fp32 matrix core:
//   out = leaky_relu( LI @ Ws + (L .* E) @ Wd )
// Block = 256 threads = 8 waves = 128 rows; wave owns a 16x128 strip.
// Weights staged transposed+interleaved in LDS: slot(2c)=Ws col c,
// slot(2c+1)=Wd col c, stride WLD=130 floats. The b1/b2 pair for a tile is
// 520 B apart -> one ds_load_2addr_b64; banks (4c+2g+khalf)%64 cover all 64
// banks exactly once -> conflict-free.
// LDS (130 KB/block) caps residency at 2 blocks/WGP = 4 waves/SIMD, so
// __launch_bounds__(256, 4) lets RA use the full per-wave VGPR budget that
// occupancy level permits (keeps all 16 B-fragments prefetchable) at zero
// occupancy cost.
// V_WMMA_F32_16X16X4_F32 layout (wave32):
//   A 16x4 : lane l (l<16): M=l, K0/K1 ; (l>=16): M=l-16, K2/K3
//   B 4x16 : lane l: N=l&15, K(2*(l>>4)) / K(2*(l>>4)+1)
//   C 16x16: lane l, vgpr v: M = v + 8*(l>>4), N = l&15
// EXEC all-1s through every WMMA (clamped loads, guarded stores only).
// ===========================================================================
__global__ void __launch_bounds__(256, 4)
gnn_fold_gemm(const float* __restrict__ LI,
              const float* __restrict__ L,
              const float* __restrict__ E,
              const float* __restrict__ Ws,
              const float* __restrict__ Wd,
              float* __restrict__ out,
              int n_ent) {
    extern __shared__ float smem[];            // 256 * WLD floats (130 KB)

    // Cooperative transposing + interleaving stage of both weight matrices.
    {
        const float4* gs = (const float4*)Ws;
        const float4* gd = (const float4*)Wd;
        for (int i = threadIdx.x; i < DIM * DIM / 4; i += 256) {
            int r  = i >> 5;                   // source row (k index)
            int c4 = (i & 31) << 2;            // source col base
            float4 ws = gs[i];
            float4 wd = gd[i];
            smem[(2 * (c4 + 0) + 0) * WLD + r] = ws.x;
            smem[(2 * (c4 + 1) + 0) * WLD + r] = ws.y;
            smem[(2 * (c4 + 2) + 0) * WLD + r] = ws.z;
            smem[(2 * (c4 + 3) + 0) * WLD + r] = ws.w;
            smem[(2 * (c4 + 0) + 1) * WLD + r] = wd.x;
            smem[(2 * (c4 + 1) + 1) * WLD + r] = wd.y;
            smem[(2 * (c4 + 2) + 1) * WLD + r] = wd.z;
            smem[(2 * (c4 + 3) + 1) * WLD + r] = wd.w;
        }
    }
    __syncthreads();

    const int wave = threadIdx.x >> 5;         // 0..7
    const int lane = threadIdx.x & 31;
    const int rowBase = blockIdx.x * 128 + wave * 16;

    const int mrow  = lane & 15;
    const int khalf = (lane >> 4) << 1;        // 0 or 2
    int row  = rowBase + mrow;
    int rowc = row < n_ent ? row : (n_ent - 1);   // clamp, keep EXEC full

    const float* liRow = LI + (size_t)rowc * DIM;
    const float* lRow  = L  + (size_t)rowc * DIM;
    const float* eRow  = E  + (size_t)rowc * DIM;

    const int col = lane & 15;
    const float* bbase = smem + (size_t)(2 * col) * WLD;   // tile n: + n*32*WLD

    v8f c[8] = {};                              // 8 N-tiles of 16x16 f32

    for (int k0 = 0; k0 < DIM; k0 += 4) {
        const int ka = k0 + khalf;              // even -> 8B aligned
        v2f a1 = *(const v2f*)(liRow + ka);     // LI_side fragment
        v2f le = *(const v2f*)(lRow + ka);
        v2f ee = *(const v2f*)(eRow + ka);
        v2f a2 = { le.x * ee.x, le.y * ee.y };  // (L_side .* ent) fragment
#pragma unroll
        for (int n = 0; n < 8; ++n) {
            const float* bp = bbase + (size_t)(n * 32) * WLD + ka;
            v2f b1 = *(const v2f*)(bp);         // Ws frag  } one
            v2f b2 = *(const v2f*)(bp + WLD);   // Wd frag  } ds_load_2addr_b64
            c[n] = __builtin_amdgcn_wmma_f32_16x16x4_f32(
                false, a1, false, b1, (short)0, c[n], false, false);
            c[n] = __builtin_amdgcn_wmma_f32_16x16x4_f32(
                false, a2, false, b2, (short)0, c[n], false, false);
        }
    }

    // Fused leaky-relu + guarded store (predication is fine post-WMMA).
    const int mOff = (lane >> 4) << 3;          // 0 or 8
#pragma unroll
    for (int n = 0; n < 8; ++n) {
#pragma unroll
        for (int v = 0; v < 8; ++v) {
            int M = rowBase + v + mOff;
            if (M < n_ent) {
                float x = c[n][v];
                out[(size_t)M * DIM + (n * 16 + col)] = (x >= 0.f) ? x : ALPHA * x;
            }
        }
    }
}

// ===========================================================================
// Host-side pipeline. CSR scratch is reused serially per matrix.
// ws layout: bufLI | bufL | counts | cursor | offs | bsums | scol | sval
// total ~= 102.4 MB + 1.2 MB + 8 MB.
// ===========================================================================
static void gnn_run_spmm(const int* rows, const int* cols, const float* vals,
                         const float* ebs, float* acc,
                         int* counts, int* cursor, int* offs, int* bsums,
                         int* scol, float* sval, hipStream_t stream) {
    const int nScan = (NBINS + SCANB - 1) / SCANB;            // 98
    gnn_zero_i<<<(2 * NBINS + 255) / 256, 256, 0, stream>>>(counts, 2 * NBINS);
    gnn_hist<<<(NNZV + 255) / 256, 256, 0, stream>>>(rows, counts, NNZV);
    gnn_scan_local<<<nScan, 256, 0, stream>>>(counts, offs, bsums, NBINS);
    gnn_scan_sums<<<1, 128, 0, stream>>>(bsums, nScan);
    gnn_scan_add<<<(NBINS + 255) / 256, 256, 0, stream>>>(offs, bsums, NBINS);
    gnn_scatter<<<(NNZV + 255) / 256, 256, 0, stream>>>(rows, cols, vals, offs,
                                                        cursor, scol, sval, NNZV);
    gnn_spmm_gather<<<(NBINS * 32 + 255) / 256, 256, 0, stream>>>(
        scol, sval, offs, counts, ebs, acc, NBINS);
}

extern "C" void kernel_launch(void* const* d_in, const int* in_sizes, int n_in,
                              void* d_out, int out_size, void* d_ws, size_t ws_size,
                              hipStream_t stream) {
    const float* ebs  = (const float*)d_in[0];
    const float* Wsu  = (const float*)d_in[1];
    const float* Wdu  = (const float*)d_in[2];
    const float* Wsi  = (const float*)d_in[3];
    const float* Wdi  = (const float*)d_in[4];
    const float* liVu = (const float*)d_in[5];
    const float* lVu  = (const float*)d_in[6];
    const float* liVi = (const float*)d_in[7];
    const float* lVi  = (const float*)d_in[8];
    const int* liRu = (const int*)d_in[9];
    const int* liCu = (const int*)d_in[10];
    const int* lRu  = (const int*)d_in[11];
    const int* lCu  = (const int*)d_in[12];
    const int* liRi = (const int*)d_in[13];
    const int* liCi = (const int*)d_in[14];
    const int* lRi  = (const int*)d_in[15];
    const int* lCi  = (const int*)d_in[16];

    float* bufLI  = (float*)d_ws;
    float* bufL   = bufLI + (size_t)NBINS * DIM;
    int*   counts = (int*)(bufL + (size_t)NBINS * DIM);
    int*   cursor = counts + NBINS;            // contiguous: one zero kernel
    int*   offs   = cursor + NBINS;
    int*   bsums  = offs + NBINS;
    int*   scol   = bsums + 128;
    float* sval   = (float*)(scol + NNZV);

    float* outU = (float*)d_out;
    float* outI = outU + (size_t)N_USERS * DIM;

    const int gemmGrd = (NBINS + 127) / 128;
    const size_t ldsB = 256 * WLD * sizeof(float);           // 130 KB

    // ---- user fold ----
    gnn_run_spmm(liRu, liCu, liVu, ebs, bufLI, counts, cursor, offs, bsums, scol, sval, stream);
    gnn_run_spmm(lRu,  lCu,  lVu,  ebs, bufL,  counts, cursor, offs, bsums, scol, sval, stream);
    gnn_fold_gemm<<<gemmGrd, 256, ldsB, stream>>>(bufLI, bufL, ebs, Wsu, Wdu, outU, N_USERS);

    // ---- item fold ----
    gnn_run_spmm(liRi, liCi, liVi, ebs, bufLI, counts, cursor, offs, bsums, scol, sval, stream);
    gnn_run_spmm(lRi,  lCi,  lVi,  ebs, bufL,  counts, cursor, offs, bsums, scol, sval, stream);
    gnn_fold_gemm<<<gemmGrd, 256, ldsB, stream>>>(bufLI, bufL,
                                                  ebs + (size_t)N_USERS * DIM,
                                                  Wsi, Wdi, outI, N_ITEMS);
}